// GCN_51677046505721
// MI455X (gfx1250) — compile-verified
//
#include <hip/hip_runtime.h>

typedef __attribute__((ext_vector_type(2))) float v2f;
typedef __attribute__((ext_vector_type(8))) float v8f;

// ---------------- degree / normalization ----------------

__global__ __launch_bounds__(256) void deg_init_k(float* deg, int n) {
    int i = blockIdx.x * blockDim.x + threadIdx.x;
    if (i < n) deg[i] = 1.0f;           // self loop contributes 1
}

__global__ __launch_bounds__(256) void deg_accum_k(const int* __restrict__ dst,
                                                   float* deg, int e) {
    int i = blockIdx.x * blockDim.x + threadIdx.x;
    if (i < e) atomicAdd(&deg[dst[i]], 1.0f);
}

__global__ __launch_bounds__(256) void rsqrt_k(float* d, int n) {
    int i = blockIdx.x * blockDim.x + threadIdx.x;
    if (i < n) d[i] = rsqrtf(d[i]);     // deg >= 1 always
}

// ---------------- fp32 WMMA GEMM: C[n x (MT*16)] = A[n x KDIM] @ B[KDIM x (MT*16)]
// One wave owns 16 rows of A and all MT column tiles (A read once, accumulators
// held in registers). Uses V_WMMA_F32_16X16X4_F32 (exact fp32).
template <int KDIM, int MT>
__global__ __launch_bounds__(256) void gemm_xw_wmma(const float* __restrict__ A,
                                                    const float* __restrict__ B,
                                                    float* __restrict__ C,
                                                    int nrows) {
    constexpr int M = MT * 16;
    int wave = (blockIdx.x * blockDim.x + threadIdx.x) >> 5;
    int lane = threadIdx.x & 31;
    int row0 = wave * 16;
    if (row0 >= nrows) return;          // wave-uniform: EXEC all-1s at WMMA

    int mr = lane & 15;                 // M index (A) / N index (B) within tile
    int kh = (lane >> 4) << 1;          // 0 or 2: K sub-offset per ISA layout

    v8f zero = {};
    v8f acc[MT];
#pragma unroll
    for (int t = 0; t < MT; ++t) acc[t] = zero;

    const float* arow = A + (size_t)(row0 + mr) * KDIM;

    for (int kk = 0; kk < KDIM; kk += 4) {
        v2f a;
        a.x = arow[kk + kh];            // A[m][kk+kh]
        a.y = arow[kk + kh + 1];        // A[m][kk+kh+1]
        const float* brow0 = B + (size_t)(kk + kh) * M;
        const float* brow1 = brow0 + M;
#pragma unroll
        for (int t = 0; t < MT; ++t) {
            v2f b;
            b.x = brow0[t * 16 + mr];   // B[kk+kh][n]
            b.y = brow1[t * 16 + mr];   // B[kk+kh+1][n]
            acc[t] = __builtin_amdgcn_wmma_f32_16x16x4_f32(
                false, a, false, b, (short)0, acc[t], false, false);
        }
    }

    // D layout: VGPR r, lanes 0-15 -> row r, lanes 16-31 -> row r+8
    int rbase = (lane >> 4) << 3;
#pragma unroll
    for (int t = 0; t < MT; ++t) {
        int col = t * 16 + mr;
#pragma unroll
        for (int r = 0; r < 8; ++r) {
            C[(size_t)(row0 + rbase + r) * M + col] = acc[t][r];
        }
    }
}

// ---------------- aggregation ----------------

// out[i][f] = bias[f] + dinv[i]^2 * h[i][f]   (self-loop term + bias)
template <int F>
__global__ __launch_bounds__(256) void agg_init_k(const float* __restrict__ h,
                                                  const float* __restrict__ dinv,
                                                  const float* __restrict__ bias,
                                                  float* __restrict__ out, int n) {
    int idx = blockIdx.x * blockDim.x + threadIdx.x;
    if (idx >= n * F) return;
    int i = idx / F;
    int f = idx - i * F;
    float di = dinv[i];
    out[idx] = bias[f] + di * di * h[idx];
}

// one wave per edge: out[dst] += dinv[src]*dinv[dst] * h[src]
template <int F>
__global__ __launch_bounds__(256) void agg_edges_k(const int* __restrict__ src,
                                                   const int* __restrict__ dst,
                                                   const float* __restrict__ dinv,
                                                   const float* __restrict__ h,
                                                   float* out, int ne) {
    int wid  = (blockIdx.x * blockDim.x + threadIdx.x) >> 5;
    int lane = threadIdx.x & 31;
    if (wid >= ne) return;
    int s = src[wid];
    int d = dst[wid];
    float nrm = dinv[s] * dinv[d];
    constexpr int PER = F / 32;         // 4 (F=128) or 2 (F=64)
    const float* hs = h + (size_t)s * F + lane * PER;
    float* o = out + (size_t)d * F + lane * PER;
    if (PER == 4) {
        float4 v = *(const float4*)hs;
        atomicAdd(o + 0, nrm * v.x);
        atomicAdd(o + 1, nrm * v.y);
        atomicAdd(o + 2, nrm * v.z);
        atomicAdd(o + 3, nrm * v.w);
    } else {
        float2 v = *(const float2*)hs;
        atomicAdd(o + 0, nrm * v.x);
        atomicAdd(o + 1, nrm * v.y);
    }
}

__global__ __launch_bounds__(256) void relu_k(float* x, int n) {
    int i = blockIdx.x * blockDim.x + threadIdx.x;
    if (i < n) x[i] = fmaxf(x[i], 0.0f);
}

// ---------------- log-softmax over 64 classes, one wave per row ----------------

__global__ __launch_bounds__(256) void log_softmax64_k(float* out, int n) {
    int wid  = (blockIdx.x * blockDim.x + threadIdx.x) >> 5;
    int lane = threadIdx.x & 31;
    if (wid >= n) return;
    float2* row = (float2*)(out + (size_t)wid * 64);
    float2 v = row[lane];
    float m = fmaxf(v.x, v.y);
#pragma unroll
    for (int off = 16; off > 0; off >>= 1) m = fmaxf(m, __shfl_xor(m, off, 32));
    float s = __expf(v.x - m) + __expf(v.y - m);
#pragma unroll
    for (int off = 16; off > 0; off >>= 1) s += __shfl_xor(s, off, 32);
    float lse = m + __logf(s);
    v.x -= lse;
    v.y -= lse;
    row[lane] = v;
}

// ---------------- launch ----------------

extern "C" void kernel_launch(void* const* d_in, const int* in_sizes, int n_in,
                              void* d_out, int out_size, void* d_ws, size_t ws_size,
                              hipStream_t stream) {
    const float* x  = (const float*)d_in[0];
    const int*  ei  = (const int*)d_in[1];
    const float* W1 = (const float*)d_in[2];
    const float* b1 = (const float*)d_in[3];
    const float* W2 = (const float*)d_in[4];
    const float* b2 = (const float*)d_in[5];

    const int N = in_sizes[0] / 128;    // 100000
    const int E = in_sizes[1] / 2;      // 1600000
    const int* srcp = ei;               // edge_index[0]
    const int* dstp = ei + E;           // edge_index[1]

    float* dinv = (float*)d_ws;                       // N floats
    float* bufA = dinv + N;                           // N*128: X@W1, reused for h1@W2
    float* bufB = bufA + (size_t)N * 128;             // N*128: aggregated layer-1
    float* out  = (float*)d_out;                      // N*64

    const int T = 256;
    // degree / dinv
    deg_init_k<<<(N + T - 1) / T, T, 0, stream>>>(dinv, N);
    deg_accum_k<<<(E + T - 1) / T, T, 0, stream>>>(dstp, dinv, E);
    rsqrt_k<<<(N + T - 1) / T, T, 0, stream>>>(dinv, N);

    // layer 1: h = X @ W1  (WMMA fp32)
    int waves1 = (N + 15) / 16;
    gemm_xw_wmma<128, 8><<<(waves1 * 32 + T - 1) / T, T, 0, stream>>>(x, W1, bufA, N);
    agg_init_k<128><<<(N * 128 + T - 1) / T, T, 0, stream>>>(bufA, dinv, b1, bufB, N);
    agg_edges_k<128><<<(E * 32 + T - 1) / T, T, 0, stream>>>(srcp, dstp, dinv, bufA, bufB, E);
    relu_k<<<(N * 128 + T - 1) / T, T, 0, stream>>>(bufB, N * 128);

    // layer 2: h2 = h1 @ W2  (WMMA fp32), aggregate directly into d_out
    gemm_xw_wmma<128, 4><<<(waves1 * 32 + T - 1) / T, T, 0, stream>>>(bufB, W2, bufA, N);
    agg_init_k<64><<<(N * 64 + T - 1) / T, T, 0, stream>>>(bufA, dinv, b2, out, N);
    agg_edges_k<64><<<(E * 32 + T - 1) / T, T, 0, stream>>>(srcp, dstp, dinv, bufA, out, E);

    // log-softmax in place
    log_softmax64_k<<<(N * 32 + T - 1) / T, T, 0, stream>>>(out, N);
}